// VectorQuantizer_20306605375791
// MI455X (gfx1250) — compile-verified
//
#include <hip/hip_runtime.h>
#include <math.h>

typedef float v2f __attribute__((ext_vector_type(2)));
typedef float v8f __attribute__((ext_vector_type(8)));

#define KCODES 8192
#define EMB    32
#define NROWS  16384
#define HW     1024        // h*w = 32*32
#define QELEMS 524288      // 16*32*32*32
#define ROWS_PER_WG 32
#define NWG    (NROWS / ROWS_PER_WG)   // 512

// ---------------- init: zero histogram, precompute |c|^2 ----------------
__global__ __launch_bounds__(256) void vq_init_kernel(const float* __restrict__ cb,
                                                      float* __restrict__ cnorm,
                                                      int* __restrict__ counts) {
    int i = blockIdx.x * 256 + threadIdx.x;   // 0..8191
    counts[i] = 0;
    const float* r = cb + i * EMB;
    float s = 0.f;
#pragma unroll
    for (int k = 0; k < EMB; k += 4) {
        float4 v = *(const float4*)(r + k);
        s += v.x * v.x + v.y * v.y + v.z * v.z + v.w * v.w;
    }
    cnorm[i] = s;
}

// ---------------- main: dual-chain WMMA distance tiles + argmin ----------------
__global__ __launch_bounds__(256) void vq_argmin_kernel(const float* __restrict__ z,
                                                        const float* __restrict__ cb,
                                                        const float* __restrict__ cnorm,
                                                        int* __restrict__ counts,
                                                        float* __restrict__ partials,
                                                        float* __restrict__ out_q,
                                                        float* __restrict__ out_tok) {
    __shared__ float s_score[8][ROWS_PER_WG];
    __shared__ int   s_idx[8][ROWS_PER_WG];
    __shared__ int   s_tok[ROWS_PER_WG];
    __shared__ float s_red[256];

    const int wg   = blockIdx.x;       // 0..511
    const int row0 = wg << 5;          // first of 32 flat rows for this WG
    const int tid  = threadIdx.x;
    const int lane = tid & 31;
    const int wave = tid >> 5;         // 0..7
    const int l16  = lane & 15;
    const int hi   = lane >> 4;        // which half of the 4-wide K chunk

    // --- two A tiles: rows [row0, row0+16) and [row0+16, row0+32) ---
    // 32-bit A layout: lane = M + 16*(K>>1), vgpr j holds K = 2*hi + j per 4-chunk.
    // z layout is [b,c,h,w]; flat row -> (b, hw), element k at stride HW.
    v2f a0[8], a1[8];
    {
        const int r0 = row0 + l16;
        const int r1 = row0 + 16 + l16;
        const float* zb0 = z + (size_t)(r0 >> 10) * (EMB * HW) + (r0 & 1023);
        const float* zb1 = z + (size_t)(r1 >> 10) * (EMB * HW) + (r1 & 1023);
#pragma unroll
        for (int kc = 0; kc < 8; ++kc) {
            const int k = kc * 4 + hi * 2;
            v2f t0; t0.x = zb0[(size_t)k * HW]; t0.y = zb0[(size_t)(k + 1) * HW];
            v2f t1; t1.x = zb1[(size_t)k * HW]; t1.y = zb1[(size_t)(k + 1) * HW];
            a0[kc] = t0; a1[kc] = t1;
        }
    }

    float best0[8], best1[8];
    int   bidx0[8], bidx1[8];
#pragma unroll
    for (int r = 0; r < 8; ++r) {
        best0[r] = 3.4e38f; bidx0[r] = 0x7fffffff;
        best1[r] = 3.4e38f; bidx1[r] = 0x7fffffff;
    }

    // --- scan this wave's 1024 codebook entries as 64 tiles of 16 ---
    const int nbase = wave << 10;
    for (int t = 0; t < 64; ++t) {
        const int entry = nbase + (t << 4) + l16;   // B layout mirrors A (N<->M)
        const float* cr = cb + (size_t)entry * EMB + hi * 2;

        v2f bv[8];
#pragma unroll
        for (int kc = 0; kc < 8; ++kc)
            bv[kc] = *(const v2f*)(cr + kc * 4);    // 8B-aligned pair (K=2*hi+{0,1})

        v8f acc0 = {}, acc1 = {};
#pragma unroll
        for (int kc = 0; kc < 8; ++kc) {            // two independent chains -> ILP
            acc0 = __builtin_amdgcn_wmma_f32_16x16x4_f32(
                false, a0[kc], false, bv[kc], (short)0, acc0, false, false);
            acc1 = __builtin_amdgcn_wmma_f32_16x16x4_f32(
                false, a1[kc], false, bv[kc], (short)0, acc1, false, false);
        }

        const float cn = cnorm[entry];
#pragma unroll
        for (int r = 0; r < 8; ++r) {               // strict < : first-index wins
            const float s0 = cn - 2.0f * acc0[r];   // within a lane (entries increase)
            const float s1 = cn - 2.0f * acc1[r];
            if (s0 < best0[r]) { best0[r] = s0; bidx0[r] = entry; }
            if (s1 < best1[r]) { best1[r] = s1; bidx1[r] = entry; }
        }
    }

    // --- cross-lane argmin (tie-break to lower index happens here, cold path) ---
#pragma unroll
    for (int mask = 1; mask < 16; mask <<= 1) {
#pragma unroll
        for (int r = 0; r < 8; ++r) {
            float os = __shfl_xor(best0[r], mask, 32);
            int   oi = __shfl_xor(bidx0[r], mask, 32);
            if (os < best0[r] || (os == best0[r] && oi < bidx0[r])) { best0[r] = os; bidx0[r] = oi; }
            os = __shfl_xor(best1[r], mask, 32);
            oi = __shfl_xor(bidx1[r], mask, 32);
            if (os < best1[r] || (os == best1[r] && oi < bidx1[r])) { best1[r] = os; bidx1[r] = oi; }
        }
    }
    if (l16 == 0) {
#pragma unroll
        for (int r = 0; r < 8; ++r) {
            const int m = r + (hi << 3);            // C layout: vgpr r -> M = r + 8*hi
            s_score[wave][m]      = best0[r];
            s_idx[wave][m]        = bidx0[r];
            s_score[wave][m + 16] = best1[r];
            s_idx[wave][m + 16]   = bidx1[r];
        }
    }
    __syncthreads();

    // --- reduce across the 8 waves; emit tokens + histogram ---
    if (tid < ROWS_PER_WG) {
        float bs = s_score[0][tid];
        int   bi = s_idx[0][tid];
#pragma unroll
        for (int w = 1; w < 8; ++w) {
            const float os = s_score[w][tid];
            const int   oi = s_idx[w][tid];
            if (os < bs || (os == bs && oi < bi)) { bs = os; bi = oi; }
        }
        s_tok[tid] = bi;
        out_tok[row0 + tid] = (float)bi;
        atomicAdd(&counts[bi], 1);
    }
    __syncthreads();

    // --- epilogue: quantized_st = z + (q - z), per-WG loss partial ---
    float part = 0.f;
#pragma unroll
    for (int p = 0; p < 4; ++p) {
        const int idx = tid + (p << 8);             // 1024 (m,ch) pairs
        const int m = idx >> 5, ch = idx & 31;
        const int row = row0 + m;
        const int bb = row >> 10, hw = row & 1023;
        const float q  = cb[(size_t)s_tok[m] * EMB + ch];
        const size_t off = (size_t)bb * (EMB * HW) + (size_t)ch * HW + hw;
        const float zv = z[off];
        const float d  = q - zv;
        out_q[off] = zv + d;                        // straight-through numerics
        part += d * d;
    }
    s_red[tid] = part;
    __syncthreads();
    for (int stg = 128; stg > 0; stg >>= 1) {
        if (tid < stg) s_red[tid] += s_red[tid + stg];
        __syncthreads();
    }
    if (tid == 0) partials[wg] = s_red[0];
}

// ---------------- finalize: loss + perplexity (deterministic trees) ----------------
__global__ __launch_bounds__(256) void vq_finalize_kernel(const float* __restrict__ partials,
                                                          const int* __restrict__ counts,
                                                          float* __restrict__ out_loss,
                                                          float* __restrict__ out_ppl) {
    __shared__ float red[256];
    const int tid = threadIdx.x;

    float s = 0.f;
    for (int i = tid; i < NWG; i += 256) s += partials[i];
    red[tid] = s;
    __syncthreads();
    for (int stg = 128; stg > 0; stg >>= 1) {
        if (tid < stg) red[tid] += red[tid + stg];
        __syncthreads();
    }
    const float loss = red[0] * (1.0f / (float)QELEMS);
    __syncthreads();

    float e = 0.f;
    for (int i = tid; i < KCODES; i += 256) {
        const float p = (float)counts[i] * (1.0f / (float)NROWS);
        e += p * logf(p + 1e-10f);
    }
    red[tid] = e;
    __syncthreads();
    for (int stg = 128; stg > 0; stg >>= 1) {
        if (tid < stg) red[tid] += red[tid + stg];
        __syncthreads();
    }
    if (tid == 0) {
        *out_loss = loss;
        *out_ppl  = expf(-red[0]);
    }
}

extern "C" void kernel_launch(void* const* d_in, const int* in_sizes, int n_in,
                              void* d_out, int out_size, void* d_ws, size_t ws_size,
                              hipStream_t stream) {
    const float* z  = (const float*)d_in[0];   // [16,32,32,32]
    const float* cb = (const float*)d_in[1];   // [8192,32]

    float* out      = (float*)d_out;
    float* out_q    = out;                     // 524288
    float* out_tok  = out + QELEMS;            // 16384
    float* out_loss = out + QELEMS + NROWS;    // 1
    float* out_ppl  = out_loss + 1;            // 1

    float* cnorm    = (float*)d_ws;                          // 8192 f32
    int*   counts   = (int*)((char*)d_ws + 32768);           // 8192 i32
    float* partials = (float*)((char*)d_ws + 65536);         // 512 f32

    vq_init_kernel<<<KCODES / 256, 256, 0, stream>>>(cb, cnorm, counts);
    vq_argmin_kernel<<<NWG, 256, 0, stream>>>(z, cb, cnorm, counts,
                                              partials, out_q, out_tok);
    vq_finalize_kernel<<<1, 256, 0, stream>>>(partials, counts, out_loss, out_ppl);
}